// GINEncoder_66340064854115
// MI455X (gfx1250) — compile-verified
//
#include <hip/hip_runtime.h>
#include <hip/hip_bf16.h>

#define NN 50000
#define NE 800000
#define D 128
#define NLAYER 3
#define BN_EPS 1e-5f

typedef __attribute__((ext_vector_type(2))) float v2f;
typedef __attribute__((ext_vector_type(8))) float v8f;

// ---------------------------------------------------------------------------
// h0 = x  (vectorized copy, float4)
// ---------------------------------------------------------------------------
__global__ void gin_copy_kernel(const float* __restrict__ in,
                                float* __restrict__ out, int n4) {
    int i = blockIdx.x * blockDim.x + threadIdx.x;
    if (i < n4) {
        ((float4*)out)[i] = ((const float4*)in)[i];
    }
}

// ---------------------------------------------------------------------------
// h0[dst, :] += x[src, :]   (2 edges per 256-thread block, 128 dims each)
// edge_index is int64, shape (2, NE): row 0 = src, row 1 = dst
// ---------------------------------------------------------------------------
__global__ void gin_edge_agg_kernel(const float* __restrict__ x,
                                    const long long* __restrict__ ei,
                                    float* __restrict__ h) {
    int e = blockIdx.x * 2 + (threadIdx.x >> 7);
    int d = threadIdx.x & (D - 1);
    if (e < NE) {
        int s = (int)ei[e];
        int t = (int)ei[NE + e];
        atomicAdd(&h[t * D + d], x[s * D + d]);
    }
}

// ---------------------------------------------------------------------------
// zero a small buffer (BN stats: sum[D], sumsq[D])
// ---------------------------------------------------------------------------
__global__ void gin_zero_kernel(float* __restrict__ p, int n) {
    int i = blockIdx.x * blockDim.x + threadIdx.x;
    if (i < n) p[i] = 0.0f;
}

// ---------------------------------------------------------------------------
// T = A(NNxD) @ W(DxD) + bias, plus per-column sum / sumsq for BatchNorm.
// One wave32 per 16x16 output tile via V_WMMA_F32_16X16X4_F32.
// Block = 256 threads = 8 waves = one 16-row panel x all 128 columns.
// Grid  = NN/16 = 3125 blocks (NN is an exact multiple of 16).
//
// Per the CDNA5 ISA f32 WMMA layouts (16x4 A, 4x16 B, 16x16 C):
//   A: lanes 0-15 hold M=lane; VGPR0/1 = K {0,1}; lanes 16-31: K {2,3}
//   B: lanes 0-15 hold N=lane; VGPR0/1 = K {0,1}; lanes 16-31: K {2,3}
//   C: VGPR v: lanes 0-15 -> (M=v, N=lane); lanes 16-31 -> (M=v+8, N=lane-16)
// ---------------------------------------------------------------------------
__global__ __launch_bounds__(256) void gin_gemm_wmma_kernel(
        const float* __restrict__ A, const float* __restrict__ W,
        const float* __restrict__ bias, float* __restrict__ T,
        float* __restrict__ stats) {
    const int wave = threadIdx.x >> 5;   // 0..7  -> column tile
    const int lane = threadIdx.x & 31;
    const int half = lane >> 4;          // 0: K+{0,1}, 1: K+{2,3}
    const int lid  = lane & 15;
    const int row0 = blockIdx.x * 16;
    const int col0 = wave * 16;

    const float* a_ptr = A + (size_t)(row0 + lid) * D + 2 * half;
    const float* b_ptr = W + (size_t)(2 * half) * D + col0 + lid;

    v8f acc = {};
#pragma unroll
    for (int k0 = 0; k0 < D; k0 += 4) {
        float2 av = *(const float2*)(a_ptr + k0);
        v2f a, b;
        a.x = av.x;                 // A[row0+lid][k0 + 2*half]
        a.y = av.y;                 // A[row0+lid][k0 + 2*half + 1]
        b.x = b_ptr[(size_t)k0 * D];        // W[k0 + 2*half][col0+lid]
        b.y = b_ptr[(size_t)(k0 + 1) * D];  // W[k0 + 2*half + 1][col0+lid]
        acc = __builtin_amdgcn_wmma_f32_16x16x4_f32(
            /*neg_a=*/false, a, /*neg_b=*/false, b,
            /*c_mod=*/(short)0, acc, /*reuse_a=*/false, /*reuse_b=*/false);
    }

    const int col = col0 + lid;
    const float bcol = bias[col];
    float s = 0.0f, sq = 0.0f;
#pragma unroll
    for (int v = 0; v < 8; ++v) {
        int row = row0 + v + 8 * half;
        float val = acc[v] + bcol;
        T[(size_t)row * D + col] = val;
        s  += val;
        sq += val * val;
    }
    // Lanes lane and lane^16 hold the same output column -> combine halves.
    s  += __shfl_xor(s, 16, 32);
    sq += __shfl_xor(sq, 16, 32);
    if (half == 0) {
        atomicAdd(&stats[col], s);
        atomicAdd(&stats[D + col], sq);
    }
}

// ---------------------------------------------------------------------------
// out = (T - mean) * rsqrt(var + eps) * gamma + beta   (+ optional ReLU)
// mean/var derived from accumulated sum / sumsq (biased variance).
// ---------------------------------------------------------------------------
__global__ void gin_bn_apply_kernel(const float* __restrict__ T,
                                    const float* __restrict__ stats,
                                    const float* __restrict__ gamma,
                                    const float* __restrict__ beta,
                                    float* __restrict__ out, int relu) {
    int i = blockIdx.x * blockDim.x + threadIdx.x;
    if (i >= NN * D) return;
    int col = i & (D - 1);
    const float invn = 1.0f / (float)NN;
    float mean = stats[col] * invn;
    float var  = stats[D + col] * invn - mean * mean;
    float rs   = rsqrtf(var + BN_EPS);
    float v    = (T[i] - mean) * rs * gamma[col] + beta[col];
    if (relu) v = fmaxf(v, 0.0f);
    out[i] = v;
}

// ---------------------------------------------------------------------------
extern "C" void kernel_launch(void* const* d_in, const int* in_sizes, int n_in,
                              void* d_out, int out_size, void* d_ws, size_t ws_size,
                              hipStream_t stream) {
    const float*     x   = (const float*)d_in[0];
    const long long* ei  = (const long long*)d_in[1];   // int64 (2, NE)
    const float*     W1  = (const float*)d_in[2];       // (L, D, D)
    const float*     b1  = (const float*)d_in[3];       // (L, D)
    const float*     g1  = (const float*)d_in[4];
    const float*     be1 = (const float*)d_in[5];
    const float*     W2  = (const float*)d_in[6];
    const float*     b2  = (const float*)d_in[7];
    const float*     g2  = (const float*)d_in[8];
    const float*     be2 = (const float*)d_in[9];

    float* out   = (float*)d_out;
    float* bufA  = (float*)d_ws;                        // h0 / hmid   (NN*D)
    float* bufB  = bufA + (size_t)NN * D;               // gemm output (NN*D)
    float* stats = bufB + (size_t)NN * D;               // sum[D], sumsq[D]

    const int ND = NN * D;
    const dim3 blk(256);
    const dim3 grid_copy((ND / 4 + 255) / 256);
    const dim3 grid_edge((NE + 1) / 2);
    const dim3 grid_zero((2 * D + 255) / 256);
    const dim3 grid_gemm(NN / 16);                      // 3125
    const dim3 grid_elem((ND + 255) / 256);

    for (int i = 0; i < NLAYER; ++i) {
        const float* xin = (i == 0) ? x : out;

        // h0 = x + sum_{j->i} x_j
        gin_copy_kernel<<<grid_copy, blk, 0, stream>>>(xin, bufA, ND / 4);
        gin_edge_agg_kernel<<<grid_edge, blk, 0, stream>>>(xin, ei, bufA);

        // h = relu(BN(h0 @ W1[i] + b1[i]))
        gin_zero_kernel<<<grid_zero, blk, 0, stream>>>(stats, 2 * D);
        gin_gemm_wmma_kernel<<<grid_gemm, blk, 0, stream>>>(
            bufA, W1 + (size_t)i * D * D, b1 + (size_t)i * D, bufB, stats);
        gin_bn_apply_kernel<<<grid_elem, blk, 0, stream>>>(
            bufB, stats, g1 + (size_t)i * D, be1 + (size_t)i * D, bufA, 1);

        // h = BN(h @ W2[i] + b2[i])  (+relu except last layer)
        gin_zero_kernel<<<grid_zero, blk, 0, stream>>>(stats, 2 * D);
        gin_gemm_wmma_kernel<<<grid_gemm, blk, 0, stream>>>(
            bufA, W2 + (size_t)i * D * D, b2 + (size_t)i * D, bufB, stats);
        gin_bn_apply_kernel<<<grid_elem, blk, 0, stream>>>(
            bufB, stats, g2 + (size_t)i * D, be2 + (size_t)i * D, out,
            (i != NLAYER - 1) ? 1 : 0);
    }
}